// TI_GATNoEdgeAttrs_23244363006445
// MI455X (gfx1250) — compile-verified
//
#include <hip/hip_runtime.h>
#include <hip/hip_bf16.h>
#include <stdint.h>

#define HH    4      // heads
#define CC    16     // per-head dim
#define HID   64     // HH*CC
#define NEGS  0.2f   // leaky relu slope
#define GG    64     // graphs
#define NCLS  10
#define NLAYERS 6

typedef __attribute__((ext_vector_type(2))) float v2f;
typedef __attribute__((ext_vector_type(8))) float v8f;

// -------- orderable-uint encoding of float for atomicMax-based segment max ----
__device__ __forceinline__ unsigned fkey(float f) {
  unsigned u = __float_as_uint(f);
  return (u & 0x80000000u) ? ~u : (u | 0x80000000u);
}
__device__ __forceinline__ float funkey(unsigned k) {
  return __uint_as_float((k & 0x80000000u) ? (k & 0x7fffffffu) : ~k);
}
#define NEG_INF_KEY 0x007fffffu   // fkey(-inf)

// =======================================================================
// Dense GEMM  h[N,64] = X[N,K] @ W[K,64]  via V_WMMA_F32_16X16X4_F32.
// Block = 128 threads = 4 waves; block owns 16 rows, wave w owns cols [16w,16w+16).
// W is staged in LDS once per block (K*64*4 <= 32 KB).
// A frag (16x4): lane<16 -> (M=lane, K+0/K+1), lane>=16 -> (M=lane-16, K+2/K+3)
// B frag (4x16): lane<16 -> (K+0/K+1, N=lane), lane>=16 -> (K+2/K+3, N=lane-16)
// D (16x16):     vgpr r, lane<16 -> (M=r, N=lane), lane>=16 -> (M=8+r, N=lane-16)
// =======================================================================
__global__ void gat_gemm_wmma(const float* __restrict__ X,
                              const float* __restrict__ W,
                              float* __restrict__ Hout,
                              int N, int K) {
  extern __shared__ float ldsW[];                   // K x 64
  const int tid = threadIdx.x;
  for (int i = tid; i < K * HID; i += 128) ldsW[i] = W[i];
  __syncthreads();

  const int lane = tid & 31;
  const int wave = tid >> 5;                        // 0..3 -> column tile
  const int m0   = blockIdx.x * 16;
  const int n0   = wave * 16;
  const int l16  = lane & 15;
  const int koff = (lane >> 4) << 1;                // 0 or 2

  int row = m0 + l16;
  if (row >= N) row = N - 1;                        // clamp: EXEC stays all-1s
  const float* __restrict__ xrow = X + (size_t)row * K;

  v8f acc = {};
  for (int k0 = 0; k0 < K; k0 += 4) {
    v2f a, b;
    a.x = xrow[k0 + koff];
    a.y = xrow[k0 + koff + 1];
    b.x = ldsW[(k0 + koff)     * HID + n0 + l16];
    b.y = ldsW[(k0 + koff + 1) * HID + n0 + l16];
    acc = __builtin_amdgcn_wmma_f32_16x16x4_f32(false, a, false, b,
                                                (short)0, acc, false, false);
  }

  const int mbase = (lane >> 4) * 8;
#pragma unroll
  for (int r = 0; r < 8; ++r) {
    const int m = m0 + mbase + r;
    if (m < N) Hout[(size_t)m * HID + n0 + l16] = acc[r];
  }
}

// es[n,h] = sum_c h[n,h,c]*a_s[h,c] ; ed likewise. One thread per (node, head).
__global__ void attn_coef(const float* __restrict__ h,
                          const float* __restrict__ a_s,
                          const float* __restrict__ a_d,
                          float* __restrict__ es, float* __restrict__ ed, int N) {
  const int i = blockIdx.x * blockDim.x + threadIdx.x;
  if (i >= N * HH) return;
  const int n = i >> 2, hh = i & 3;
  const float* hv = h + (size_t)n * HID + hh * CC;
  float s0 = 0.f, s1 = 0.f;
#pragma unroll
  for (int c = 0; c < CC; ++c) {
    const float v = hv[c];
    s0 += v * a_s[hh * CC + c];
    s1 += v * a_d[hh * CC + c];
  }
  es[i] = s0;
  ed[i] = s1;
}

// Reset per-layer accumulators: mkey=-inf key, denom=0, out=0 (float4 stores).
__global__ void layer_init(unsigned* __restrict__ mkey, float* __restrict__ denom,
                           float4* __restrict__ out4, int N) {
  const int i = blockIdx.x * blockDim.x + threadIdx.x;
  if (i < N * HH) { mkey[i] = NEG_INF_KEY; denom[i] = 0.f; }
  if (i < N * (HID / 4)) out4[i] = make_float4(0.f, 0.f, 0.f, 0.f);
}

__device__ __forceinline__ void edge_nodes(int eid, int E,
                                           const int* __restrict__ src,
                                           const int* __restrict__ dst,
                                           int& s, int& d) {
  if (eid < E) { s = src[eid]; d = dst[eid]; }
  else         { s = d = eid - E; }               // implicit self-loop
}

// Pass 1: e = leakyrelu(es[s]+ed[d]); stash e in ee[]; segment max by dst.
__global__ void edge_max(const int* __restrict__ src, const int* __restrict__ dst,
                         const float* __restrict__ es, const float* __restrict__ ed,
                         unsigned* __restrict__ mkey, float* __restrict__ ee,
                         int E, int Etot) {
  const int i = blockIdx.x * blockDim.x + threadIdx.x;
  if (i >= Etot * HH) return;
  const int eid = i >> 2, hh = i & 3;
  int s, d; edge_nodes(eid, E, src, dst, s, d);
  float e = es[s * HH + hh] + ed[d * HH + hh];
  e = e > 0.f ? e : NEGS * e;
  ee[i] = e;                                      // reuse buffer: raw e for pass 2
  atomicMax(&mkey[d * HH + hh], fkey(e));
}

// Pass 2: ee = exp(e - m[dst]) (in place); denom[dst] += ee.
__global__ void edge_expsum(const int* __restrict__ src, const int* __restrict__ dst,
                            const unsigned* __restrict__ mkey,
                            float* __restrict__ ee, float* __restrict__ denom,
                            int E, int Etot) {
  const int i = blockIdx.x * blockDim.x + threadIdx.x;
  if (i >= Etot * HH) return;
  const int eid = i >> 2, hh = i & 3;
  int s, d; edge_nodes(eid, E, src, dst, s, d);
  const float v = __expf(ee[i] - funkey(mkey[d * HH + hh]));
  ee[i] = v;
  atomicAdd(&denom[d * HH + hh], v);
}

// Pass 3: out[dst,f] += h[src,f] * (ee / denom[dst,head(f)]).
// One thread per (edge, 4-feature quad): b128 load of h + 4 atomic f32 adds.
__global__ void edge_aggr(const int* __restrict__ src, const int* __restrict__ dst,
                          const float* __restrict__ h, const float* __restrict__ ee,
                          const float* __restrict__ denom, float* __restrict__ out,
                          int E, int Etot) {
  const long i = (long)blockIdx.x * blockDim.x + threadIdx.x;
  if (i >= (long)Etot * (HID / 4)) return;
  const int eid = (int)(i >> 4);
  const int q   = (int)(i & 15);                  // quad index; head = q/4
  const int f0  = q << 2;
  const int hh  = q >> 2;
  int s, d; edge_nodes(eid, E, src, dst, s, d);
  const float alpha = ee[eid * HH + hh] / denom[d * HH + hh];
  const float4 hv = *(const float4*)(h + (size_t)s * HID + f0);
  float* o = out + (size_t)d * HID + f0;
  atomicAdd(o + 0, hv.x * alpha);
  atomicAdd(o + 1, hv.y * alpha);
  atomicAdd(o + 2, hv.z * alpha);
  atomicAdd(o + 3, hv.w * alpha);
}

// x = relu(x + bias), float4-wide
__global__ void bias_relu(float4* __restrict__ x4, const float4* __restrict__ b4, int N) {
  const int i = blockIdx.x * blockDim.x + threadIdx.x;
  if (i >= N * (HID / 4)) return;
  float4 v = x4[i];
  const float4 b = b4[i & 15];
  v.x = fmaxf(v.x + b.x, 0.f);
  v.y = fmaxf(v.y + b.y, 0.f);
  v.z = fmaxf(v.z + b.z, 0.f);
  v.w = fmaxf(v.w + b.w, 0.f);
  x4[i] = v;
}

__global__ void pool_init(unsigned* __restrict__ gkey) {
  const int i = blockIdx.x * blockDim.x + threadIdx.x;
  if (i < GG * HID) gkey[i] = NEG_INF_KEY;
}

// global max pool over graphs
__global__ void pool_max(const float* __restrict__ x, const int* __restrict__ batch,
                         unsigned* __restrict__ gkey, int N) {
  const int i = blockIdx.x * blockDim.x + threadIdx.x;
  if (i >= N * HID) return;
  const int n = i >> 6, f = i & 63;
  atomicMax(&gkey[batch[n] * HID + f], fkey(x[i]));
}

// MLP head: out[G,10] = relu(g @ fc1 + b1) @ fc2 + b2. One thread per graph.
__global__ void mlp_head(const unsigned* __restrict__ gkey,
                         const float* __restrict__ fc1w, const float* __restrict__ fc1b,
                         const float* __restrict__ fc2w, const float* __restrict__ fc2b,
                         float* __restrict__ out) {
  const int t = threadIdx.x;
  if (t >= GG) return;
  float hid[2 * CC];
#pragma unroll
  for (int o = 0; o < 2 * CC; ++o) {
    float s = fc1b[o];
    for (int j = 0; j < HID; ++j)
      s += funkey(gkey[t * HID + j]) * fc1w[j * (2 * CC) + o];
    hid[o] = s > 0.f ? s : 0.f;
  }
#pragma unroll
  for (int o = 0; o < NCLS; ++o) {
    float s = fc2b[o];
#pragma unroll
    for (int j = 0; j < 2 * CC; ++j) s += hid[j] * fc2w[j * NCLS + o];
    out[t * NCLS + o] = s;
  }
}

// =======================================================================
extern "C" void kernel_launch(void* const* d_in, const int* in_sizes, int n_in,
                              void* d_out, int out_size, void* d_ws, size_t ws_size,
                              hipStream_t stream) {
  const float* x          = (const float*)d_in[0];
  const int*   src        = (const int*)d_in[1];
  const int*   dst        = (const int*)d_in[2];
  const int*   batch      = (const int*)d_in[3];
  const float* W0         = (const float*)d_in[4];
  const float* a_src0     = (const float*)d_in[5];
  const float* a_dst0     = (const float*)d_in[6];
  const float* b0         = (const float*)d_in[7];
  const float* W_rest     = (const float*)d_in[8];
  const float* a_src_rest = (const float*)d_in[9];
  const float* a_dst_rest = (const float*)d_in[10];
  const float* b_rest     = (const float*)d_in[11];
  const float* fc1_w      = (const float*)d_in[12];
  const float* fc1_b      = (const float*)d_in[13];
  const float* fc2_w      = (const float*)d_in[14];
  const float* fc2_b      = (const float*)d_in[15];

  const int N    = in_sizes[3];         // batch vector length == #nodes
  const int E    = in_sizes[1];
  const int Etot = E + N;               // + self loops
  const int F_IN = in_sizes[0] / N;     // 128

  float* ws = (float*)d_ws;
  float*    xbuf  = ws;                ws += (size_t)N * HID;
  float*    hbuf  = ws;                ws += (size_t)N * HID;
  float*    es    = ws;                ws += (size_t)N * HH;
  float*    ed    = ws;                ws += (size_t)N * HH;
  float*    denom = ws;                ws += (size_t)N * HH;
  unsigned* mkey  = (unsigned*)ws;     ws += (size_t)N * HH;
  float*    ee    = ws;                ws += (size_t)Etot * HH;
  unsigned* gkey  = (unsigned*)ws;     ws += (size_t)GG * HID;

  const int T = 256;
  const unsigned nh_blk  = (unsigned)((N * HH + T - 1) / T);
  const unsigned nf_blk  = (unsigned)(((size_t)N * HID + T - 1) / T);
  const unsigned nq_blk  = (unsigned)(((size_t)N * (HID / 4) + T - 1) / T);
  const unsigned eh_blk  = (unsigned)(((size_t)Etot * HH + T - 1) / T);
  const unsigned eq_blk  = (unsigned)(((size_t)Etot * (HID / 4) + T - 1) / T);

  const float* xin = x;
  int K = F_IN;
  for (int layer = 0; layer < NLAYERS; ++layer) {
    const float* W  = layer == 0 ? W0     : W_rest     + (size_t)(layer - 1) * HID * HID;
    const float* as = layer == 0 ? a_src0 : a_src_rest + (size_t)(layer - 1) * HID;
    const float* ad = layer == 0 ? a_dst0 : a_dst_rest + (size_t)(layer - 1) * HID;
    const float* bb = layer == 0 ? b0     : b_rest     + (size_t)(layer - 1) * HID;

    gat_gemm_wmma<<<(N + 15) / 16, 128, (size_t)K * HID * sizeof(float), stream>>>(
        xin, W, hbuf, N, K);
    attn_coef<<<nh_blk, T, 0, stream>>>(hbuf, as, ad, es, ed, N);
    // safe to reset xbuf here: the GEMM above has already consumed it
    layer_init<<<nq_blk, T, 0, stream>>>(mkey, denom, (float4*)xbuf, N);
    edge_max<<<eh_blk, T, 0, stream>>>(src, dst, es, ed, mkey, ee, E, Etot);
    edge_expsum<<<eh_blk, T, 0, stream>>>(src, dst, mkey, ee, denom, E, Etot);
    edge_aggr<<<eq_blk, T, 0, stream>>>(src, dst, hbuf, ee, denom, xbuf, E, Etot);
    bias_relu<<<nq_blk, T, 0, stream>>>((float4*)xbuf, (const float4*)bb, N);
    xin = xbuf;
    K = HID;
  }

  pool_init<<<(GG * HID + T - 1) / T, T, 0, stream>>>(gkey);
  pool_max<<<nf_blk, T, 0, stream>>>(xbuf, batch, gkey, N);
  mlp_head<<<1, GG, 0, stream>>>(gkey, fc1_w, fc1_b, fc2_w, fc2_b, (float*)d_out);
}